// ArapEnergyHessianAnalytical_46059229282944
// MI455X (gfx1250) — compile-verified
//
#include <hip/hip_runtime.h>

typedef __attribute__((ext_vector_type(2))) float v2f;
typedef __attribute__((ext_vector_type(8))) float v8f;

#define NPTS 5000
#define KNBR 20
#define NZDIM 128
#define HID 1024
#define M3 15000          // 3*N columns of the Jacobian
#define QM 15008          // padded to 938 * 16
#define NTILE 938
#define DPAD 5008         // >= (QM-1)/3 + 1
#define KCH (QM / 8)      // 1876, K-chunk for GEMM2 split
#define HSCALE (2.0f / (float)(NPTS * KNBR))

static __device__ __forceinline__ v8f wmma_f32(v2f a, v2f b, v8f c) {
  // V_WMMA_F32_16X16X4_F32 : D = A(16x4,f32) * B(4x16,f32) + C(16x16,f32)
  return __builtin_amdgcn_wmma_f32_16x16x4_f32(false, a, false, b, (short)0, c,
                                               false, false);
}

// ---------------- utility ----------------
__global__ void k_zero(float* __restrict__ p, int n) {
  int i = blockIdx.x * blockDim.x + threadIdx.x;
  if (i < n) p[i] = 0.f;
}

// ---------------- ReLU activation pattern: s_j = (c.W1 + b1 > 0) ----------------
__global__ void k_mask(const float* __restrict__ code, const float* __restrict__ W1,
                       const float* __restrict__ b1, float* __restrict__ s) {
  int j = blockIdx.x * blockDim.x + threadIdx.x;
  if (j >= HID) return;
  float a = b1[j];
  for (int i = 0; i < NZDIM; ++i) a += code[i] * W1[i * HID + j];
  s[j] = (a > 0.f) ? 1.f : 0.f;
}

// ---------------- Apre = W1 .* s (mask columns) ----------------
__global__ void k_aprep(const float* __restrict__ W1, const float* __restrict__ s,
                        float* __restrict__ Apre) {
  int t = blockIdx.x * blockDim.x + threadIdx.x;
  if (t < NZDIM * HID) Apre[t] = W1[t] * s[t % HID];
}

// ---------------- GEMM1: Q[128 x QM] = Apre[128 x 1024] @ W2[1024 x 15000] ----------------
// One wave per (32-row x 16-col) output: two 16x16 WMMA tiles sharing each B load.
// Branchless tail: pad columns load a clamped address and are zeroed by a 0/1 mask.
__global__ void __launch_bounds__(32) k_gemm1(const float* __restrict__ Apre,
                                              const float* __restrict__ W2,
                                              float* __restrict__ Q) {
  const int tn = blockIdx.x;            // 0..937 column tile
  const int tg = blockIdx.y;            // 0..3   group of two row tiles
  const int lane = threadIdx.x;
  const int l15 = lane & 15;
  const int kh = (lane >> 4) * 2;       // lanes 0-15: K=k0,k0+1 ; lanes 16-31: K=k0+2,k0+3
  const int mrow0 = (tg * 2 + 0) * 16 + l15;
  const int mrow1 = (tg * 2 + 1) * 16 + l15;
  const int ncol = tn * 16 + l15;
  const float bm = (ncol < M3) ? 1.f : 0.f;
  const int nc = (ncol < M3) ? ncol : 0;
  const float* __restrict__ a0p = Apre + mrow0 * HID + kh;
  const float* __restrict__ a1p = Apre + mrow1 * HID + kh;
  const float* __restrict__ bp = W2 + nc + (size_t)kh * M3;
  v8f acc0 = {}, acc1 = {};
  for (int k0 = 0; k0 < HID; k0 += 4) {
    v2f b;
    b.x = bp[(size_t)k0 * M3] * bm;
    b.y = bp[(size_t)(k0 + 1) * M3] * bm;
    v2f a;
    a.x = a0p[k0];
    a.y = a0p[k0 + 1];
    acc0 = wmma_f32(a, b, acc0);
    a.x = a1p[k0];
    a.y = a1p[k0 + 1];
    acc1 = wmma_f32(a, b, acc1);
  }
  const int col = tn * 16 + l15;
  const int rbase0 = (tg * 2 + 0) * 16 + 8 * (lane >> 4);
  const int rbase1 = (tg * 2 + 1) * 16 + 8 * (lane >> 4);
#pragma unroll
  for (int r = 0; r < 8; ++r)
    Q[(size_t)(rbase0 + r) * QM + col] = acc0[r];
#pragma unroll
  for (int r = 0; r < 8; ++r)
    Q[(size_t)(rbase1 + r) * QM + col] = acc1[r];
}

// ---------------- weighted degrees: d[n] = sum_out w + sum_in w ----------------
__global__ void k_degree(const float* __restrict__ wmat, const int* __restrict__ nbr,
                         const int* __restrict__ numN, float* __restrict__ d) {
  int n = blockIdx.x * blockDim.x + threadIdx.x;
  if (n >= NPTS) return;
  const int nn = numN[n];
  float dsum = 0.f;
  for (int k = 0; k < KNBR; ++k) {
    const float w = (k < nn) ? wmat[n * KNBR + k] : 0.f;
    dsum += w;
    if (w != 0.f) atomicAdd(&d[nbr[n * KNBR + k]], w);
  }
  atomicAdd(&d[n], dsum);
}

// ---------------- dexp[m] = d[m/3] ----------------
__global__ void k_dexpand(const float* __restrict__ d, float* __restrict__ dexp) {
  int m = blockIdx.x * blockDim.x + threadIdx.x;
  if (m < QM) dexp[m] = d[m / 3];
}

// ---------------- Z[:, 3n+c] = sum_k w_{nk} * Q[:, 3*j(n,k)+c]  (pure gather) ----------------
__global__ void __launch_bounds__(128) k_gather(const float* __restrict__ Q,
                                                const int* __restrict__ nbr,
                                                const float* __restrict__ wmat,
                                                const int* __restrict__ numN,
                                                float* __restrict__ Z) {
  const int n = blockIdx.x;
  const int i = threadIdx.x;            // latent row 0..127
  __shared__ int sj[KNBR];
  __shared__ float sw[KNBR];
  if (i < KNBR) {
    const int nn = numN[n];
    sj[i] = nbr[n * KNBR + i] * 3;
    sw[i] = (i < nn) ? wmat[n * KNBR + i] : 0.f;
  }
  __syncthreads();
  const float* qrow = Q + (size_t)i * QM;
  float a0 = 0.f, a1 = 0.f, a2 = 0.f;
#pragma unroll
  for (int k = 0; k < KNBR; ++k) {
    const float w = sw[k];
    const int j3 = sj[k];
    a0 += w * qrow[j3];
    a1 += w * qrow[j3 + 1];
    a2 += w * qrow[j3 + 2];
  }
  const size_t base = (size_t)i * QM + n * 3;
  Z[base] = a0; Z[base + 1] = a1; Z[base + 2] = a2;
}

// ---------------- G[i,m] = dexp[m]*Q[i,m] - Z[i,m] ----------------
__global__ void k_gform(const float* __restrict__ Q, const float* __restrict__ Z,
                        const float* __restrict__ dexp, float* __restrict__ G) {
  const int i = blockIdx.y;
  const int m = blockIdx.x * blockDim.x + threadIdx.x;
  if (m < QM) {
    const size_t t = (size_t)i * QM + m;
    G[t] = dexp[m] * Q[t] - Z[t];
  }
}

// ---------------- GEMM2: H += sc * A Bᵀ   (A,B row-major [128 x QM]) ----------------
// grid (8, 8, 8): K split into 8 chunks, atomic reduce into H.
__global__ void __launch_bounds__(32) k_gemm2(const float* __restrict__ A,
                                              const float* __restrict__ B,
                                              float* __restrict__ H, float sc) {
  const int ti = blockIdx.x;
  const int tj = blockIdx.y;
  const int chunk = blockIdx.z;
  const int lane = threadIdx.x;
  const int l15 = lane & 15;
  const int kh = (lane >> 4) * 2;
  const float* __restrict__ ap = A + (size_t)(ti * 16 + l15) * QM + chunk * KCH + kh;
  const float* __restrict__ bp = B + (size_t)(tj * 16 + l15) * QM + chunk * KCH + kh;
  v8f acc = {};
  for (int st = 0; st < KCH; st += 4) {
    v2f a, b;
    a.x = ap[st];
    a.y = ap[st + 1];
    b.x = bp[st];
    b.y = bp[st + 1];
    acc = wmma_f32(a, b, acc);
  }
#pragma unroll
  for (int r = 0; r < 8; ++r) {
    const int row = ti * 16 + r + 8 * (lane >> 4);
    const int col = tj * 16 + l15;
    atomicAdd(&H[row * NZDIM + col], sc * acc[r]);
  }
}

extern "C" void kernel_launch(void* const* d_in, const int* in_sizes, int n_in,
                              void* d_out, int out_size, void* d_ws, size_t ws_size,
                              hipStream_t stream) {
  (void)in_sizes; (void)n_in; (void)out_size; (void)ws_size;
  const float* code = (const float*)d_in[0];   // [1,128]
  // d_in[1] = xyz1 : only affects the linear term, not the Hessian
  const float* wmat = (const float*)d_in[2];   // [N,K]
  const float* W1   = (const float*)d_in[3];   // [128,1024]
  const float* b1   = (const float*)d_in[4];   // [1024]
  const float* W2   = (const float*)d_in[5];   // [1024,15000]
  // d_in[6] = b2 : does not affect the Hessian
  const int* nbr    = (const int*)d_in[7];     // [N,K]
  const int* numN   = (const int*)d_in[8];     // [N]
  float* H  = (float*)d_out;                   // [128,128]
  float* ws = (float*)d_ws;

  float* s    = ws;                               // 1024
  float* Apre = s + HID;                          // 128*1024
  float* Q    = Apre + NZDIM * HID;               // 128*QM
  float* Z    = Q + (size_t)NZDIM * QM;           // 128*QM
  float* dd   = Z + (size_t)NZDIM * QM;           // DPAD
  float* dexp = dd + DPAD;                        // QM
  float* G    = dexp + QM;                        // 128*QM

  const int ZN = NZDIM * QM;
  k_zero<<<(ZN + 255) / 256, 256, 0, stream>>>(Z, ZN);
  k_zero<<<(DPAD + 255) / 256, 256, 0, stream>>>(dd, DPAD);
  k_zero<<<(NZDIM * NZDIM + 255) / 256, 256, 0, stream>>>(H, NZDIM * NZDIM);

  k_mask<<<(HID + 255) / 256, 256, 0, stream>>>(code, W1, b1, s);
  k_aprep<<<(NZDIM * HID + 255) / 256, 256, 0, stream>>>(W1, s, Apre);

  dim3 g1(NTILE, NZDIM / 32);
  k_gemm1<<<g1, 32, 0, stream>>>(Apre, W2, Q);

  k_degree<<<(NPTS + 255) / 256, 256, 0, stream>>>(wmat, nbr, numN, dd);
  k_dexpand<<<(QM + 255) / 256, 256, 0, stream>>>(dd, dexp);
  k_gather<<<NPTS, 128, 0, stream>>>(Q, nbr, wmat, numN, Z);

  dim3 gg((QM + 255) / 256, NZDIM);
  k_gform<<<gg, 256, 0, stream>>>(Q, Z, dexp, G);

  dim3 g2(NZDIM / 16, NZDIM / 16, 8);
  k_gemm2<<<g2, 32, 0, stream>>>(Q, G, H, HSCALE);
  k_gemm2<<<g2, 32, 0, stream>>>(Z, Q, H, -HSCALE);
}